// MRL_22668837388856
// MI455X (gfx1250) — compile-verified
//
#include <hip/hip_runtime.h>

#ifndef __has_builtin
#define __has_builtin(x) 0
#endif

#if __has_builtin(__builtin_amdgcn_global_load_async_to_lds_b128)
#define USE_ASYNC_LDS 1
#else
#define USE_ASYNC_LDS 0
#endif

#define N_LEVELS    16
#define LDS_LEVELS  8
#define LDS_ENTRIES 4144   /* = offsets[8] = 16*(2^8 - 1) + 8*8 entries (float4 each) */
#define OUT_STRIDE  65     /* 1 + 16*4 floats per point */
#define BLOCK       256    /* 8 wave32 waves */

// The async-to-LDS builtin takes typed v4i pointers: (global src, LDS dst, imm offset, imm cpol).
typedef int v4i __attribute__((vector_size(16)));
typedef __attribute__((address_space(1))) v4i* as1_v4i_p;
typedef __attribute__((address_space(3))) v4i* as3_v4i_p;

__global__ __launch_bounds__(BLOCK) void mrl_grid1d_kernel(
    const float*  __restrict__ x,
    const float4* __restrict__ data,     // [T,4] table, 16B-aligned rows
    const float*  __restrict__ scales,   // [16]
    const int*    __restrict__ offsets,  // [16]
    float*        __restrict__ out,      // [N,65]
    int n_points)
{
    // Hot prefix of the table (levels 0..7) cached in LDS: 66,304 B static LDS.
    __shared__ float4 stab[LDS_ENTRIES];

#if USE_ASYNC_LDS
    // CDNA5 async global->LDS copy (ASYNCcnt-tracked), b128 per lane.
    for (int i = threadIdx.x; i < LDS_ENTRIES; i += BLOCK) {
        __builtin_amdgcn_global_load_async_to_lds_b128(
            (as1_v4i_p)(data + i), (as3_v4i_p)(stab + i), /*offset=*/0, /*cpol=*/0);
    }
#if __has_builtin(__builtin_amdgcn_s_wait_asynccnt)
    __builtin_amdgcn_s_wait_asynccnt(0);
#else
    asm volatile("s_wait_asynccnt 0" ::: "memory");
#endif
#else
    for (int i = threadIdx.x; i < LDS_ENTRIES; i += BLOCK)
        stab[i] = data[i];
#endif
    __syncthreads();

    const int n = blockIdx.x * BLOCK + threadIdx.x;
    if (n >= n_points) return;

    const float xc = fminf(fmaxf(x[n], 0.0f), 1.0f);
    float* __restrict__ orow = out + (size_t)n * OUT_STRIDE;
    // Output is a 136MB write-once stream: NT stores keep the 16.8MB table L2-resident.
    __builtin_nontemporal_store(xc, orow);

    // Coarse levels: gather from LDS (ds_load_b128).
#pragma unroll
    for (int l = 0; l < LDS_LEVELS; ++l) {
        const float s    = scales[l];     // uniform -> s_load
        const int   base = offsets[l];    // uniform -> s_load
        const float fx = xc * s;
        const float fi = floorf(fx);
        const int   i0 = (int)fi;
        const float f  = fx - fi;
        const float w0 = 1.0f - f;
        const float4 v0 = stab[base + i0];
        const float4 v1 = stab[base + i0 + 1];
        float* p = orow + 1 + 4 * l;
        __builtin_nontemporal_store(w0 * v0.x + f * v1.x, p + 0);
        __builtin_nontemporal_store(w0 * v0.y + f * v1.y, p + 1);
        __builtin_nontemporal_store(w0 * v0.z + f * v1.z, p + 2);
        __builtin_nontemporal_store(w0 * v0.w + f * v1.w, p + 3);
    }

    // Fine levels: gather from global (global_load_b128, L2-resident table).
#pragma unroll
    for (int l = LDS_LEVELS; l < N_LEVELS; ++l) {
        const float s    = scales[l];
        const int   base = offsets[l];
        const float fx = xc * s;
        const float fi = floorf(fx);
        const int   i0 = (int)fi;
        const float f  = fx - fi;
        const float w0 = 1.0f - f;
        const float4 v0 = data[base + i0];
        const float4 v1 = data[base + i0 + 1];
        float* p = orow + 1 + 4 * l;
        __builtin_nontemporal_store(w0 * v0.x + f * v1.x, p + 0);
        __builtin_nontemporal_store(w0 * v0.y + f * v1.y, p + 1);
        __builtin_nontemporal_store(w0 * v0.z + f * v1.z, p + 2);
        __builtin_nontemporal_store(w0 * v0.w + f * v1.w, p + 3);
    }
}

extern "C" void kernel_launch(void* const* d_in, const int* in_sizes, int n_in,
                              void* d_out, int out_size, void* d_ws, size_t ws_size,
                              hipStream_t stream) {
    (void)n_in; (void)out_size; (void)d_ws; (void)ws_size;
    const float*  x       = (const float*) d_in[0];
    const float4* data    = (const float4*)d_in[1];
    const float*  scales  = (const float*) d_in[2];
    const int*    offsets = (const int*)   d_in[3];
    float*        out     = (float*)d_out;
    const int n_points = in_sizes[0];          // x is [N,1]
    const int grid = (n_points + BLOCK - 1) / BLOCK;
    mrl_grid1d_kernel<<<grid, BLOCK, 0, stream>>>(x, data, scales, offsets, out, n_points);
}